// GCN_15479062135291
// MI455X (gfx1250) — compile-verified
//
#include <hip/hip_runtime.h>

typedef __attribute__((ext_vector_type(2))) float v2f;
typedef __attribute__((ext_vector_type(8))) float v8f;

// Align element counts to 64 floats (256 bytes).
static inline size_t ws_align(size_t n) { return (n + 63) & ~(size_t)63; }

// ---------------------------------------------------------------------------
// Elementwise zero (grid-stride).
__global__ void zero_f32(float* __restrict__ p, size_t n) {
  size_t i = (size_t)blockIdx.x * blockDim.x + threadIdx.x;
  size_t stride = (size_t)gridDim.x * blockDim.x;
  for (; i < n; i += stride) p[i] = 0.0f;
}

// ---------------------------------------------------------------------------
// Degree count over destination indices (self-loops handled in deg_to_dinv).
__global__ void deg_count(const int* __restrict__ dst, float* __restrict__ deg,
                          int E) {
  int i = blockIdx.x * blockDim.x + threadIdx.x;
  if (i < E) atomicAdd(&deg[dst[i]], 1.0f);
}

__global__ void deg_to_dinv(float* __restrict__ deg, int N) {
  int i = blockIdx.x * blockDim.x + threadIdx.x;
  if (i < N) deg[i] = rsqrtf(deg[i] + 1.0f);  // +1 self-loop => always > 0
}

// ---------------------------------------------------------------------------
// Zero-pad W[K,Hout] -> Wp[K,Hpad] and bias[Hout] -> bp[Hpad] so the GEMM
// inner loop needs no column predicates (no EXEC divergence around loads).
__global__ void pad_weights(const float* __restrict__ W,
                            const float* __restrict__ b,
                            float* __restrict__ Wp, float* __restrict__ bp,
                            int K, int Hout, int Hpad) {
  int i = blockIdx.x * blockDim.x + threadIdx.x;
  int total = K * Hpad;
  if (i < total) {
    int r = i / Hpad, c = i - r * Hpad;
    Wp[i] = (c < Hout) ? W[r * Hout + c] : 0.0f;
  }
  if (i < Hpad) bp[i] = (i < Hout) ? b[i] : 0.0f;
}

// ---------------------------------------------------------------------------
// out[N,Hout] = A[N,K] @ W[K,Hpad] + bias, one 16x16 tile per wave32 via
// V_WMMA_F32_16X16X4_F32. Branchless inner loop:
//   - A row index clamped to N-1 (garbage rows masked at the store),
//   - W/bias pre-padded to Hpad (multiple of 16), so every load is in-bounds.
//
// Fragment layouts (ISA 7.12.2):
//   A (16x4): lanes 0-15 -> K pair {k,k+1}, lanes 16-31 -> {k+2,k+3}; M=lane&15.
//   B (4x16): mirrored (rows striped across lanes within a VGPR).
//   C/D: 8 VGPRs; lane<16 -> M=v, lane>=16 -> M=8+v; N=lane&15.
__global__ void gemm_wmma_f32(const float* __restrict__ A,
                              const float* __restrict__ W,
                              const float* __restrict__ bias,
                              float* __restrict__ out,
                              int N, int K, int Hout, int Hpad) {
  const int tileN = blockIdx.x;  // column tile (Hpad/16)
  const int tileM = blockIdx.y;  // row tile    (ceil(N/16))
  const int lane  = threadIdx.x & 31;
  const int half  = lane >> 4;   // 0: K pair {0,1}; 1: K pair {2,3}
  const int l     = lane & 15;

  const int rowA  = tileM * 16 + l;
  const int rowLd = rowA < N ? rowA : (N - 1);  // clamp: loads always valid
  const int colB  = tileN * 16 + l;             // < Hpad by construction

  const float* __restrict__ Ap = A + (size_t)rowLd * K + half * 2;
  const float* __restrict__ Bp = W + (size_t)(half * 2) * Hpad + colB;

  v8f c = {};
#pragma unroll 8
  for (int k = 0; k < K; k += 4) {
    v2f a = *(const v2f*)(Ap + k);        // global_load_b64, K pair for this half
    v2f b;
    b.x = Bp[(size_t)k * Hpad];           // row k + half*2
    b.y = Bp[(size_t)k * Hpad + Hpad];    // row k + half*2 + 1
    // 8 args: (neg_a, A, neg_b, B, c_mod, C, reuse_a, reuse_b)
    c = __builtin_amdgcn_wmma_f32_16x16x4_f32(false, a, false, b,
                                              (short)0, c, false, false);
  }

  const bool colOK = colB < Hout;
  const float bv = bias[colB];
#pragma unroll
  for (int v = 0; v < 8; ++v) {
    const int row = tileM * 16 + half * 8 + v;
    if (row < N && colOK) out[(size_t)row * Hout + colB] = c[v] + bv;
  }
}

// ---------------------------------------------------------------------------
// One wave32 per edge: acc[dst] += t[src] * dinv[src] * dinv[dst].
// Lanes stride the feature dimension -> coalesced loads + coalesced atomics.
__global__ void scatter_edges(const float* __restrict__ t,
                              const int* __restrict__ src,
                              const int* __restrict__ dst,
                              const float* __restrict__ dinv,
                              float* __restrict__ acc,
                              int E, int Hdim) {
  const int e = blockIdx.x * (blockDim.x >> 5) + (threadIdx.x >> 5);
  if (e >= E) return;
  const int lane = threadIdx.x & 31;
  const int s = src[e];
  const int d = dst[e];
  const float nrm = dinv[s] * dinv[d];
  const float* __restrict__ tp = t + (size_t)s * Hdim;
  float* __restrict__ ap = acc + (size_t)d * Hdim;
  for (int f = lane; f < Hdim; f += 32)
    atomicAdd(&ap[f], tp[f] * nrm);
}

// ---------------------------------------------------------------------------
// Fused self-loop contribution + optional ReLU:
//   acc[i,f] += t[i,f] * dinv[i]^2 ; acc = relu(acc) if requested.
__global__ void finalize_selfloop(const float* __restrict__ t,
                                  const float* __restrict__ dinv,
                                  float* __restrict__ acc,
                                  int N, int Hdim, int do_relu) {
  const size_t idx = (size_t)blockIdx.x * blockDim.x + threadIdx.x;
  const size_t total = (size_t)N * (size_t)Hdim;
  if (idx >= total) return;
  const int row = (int)(idx / (size_t)Hdim);
  const float di = dinv[row];
  float v = acc[idx] + t[idx] * di * di;
  acc[idx] = do_relu ? fmaxf(v, 0.0f) : v;
}

// ---------------------------------------------------------------------------
extern "C" void kernel_launch(void* const* d_in, const int* in_sizes, int n_in,
                              void* d_out, int out_size, void* d_ws,
                              size_t ws_size, hipStream_t stream) {
  const float* x  = (const float*)d_in[0];
  const int*   ei = (const int*)d_in[1];   // [2, E]: row0 = src, row1 = dst
  const float* W1 = (const float*)d_in[2];
  const float* b1 = (const float*)d_in[3];
  const float* W2 = (const float*)d_in[4];
  const float* b2 = (const float*)d_in[5];
  const float* W3 = (const float*)d_in[6];
  const float* b3 = (const float*)d_in[7];

  const int F = 128;
  const int N = in_sizes[0] / F;
  const int E = in_sizes[1] / 2;
  const int H = in_sizes[3];   // 128
  const int C = in_sizes[7];   // 40
  const int Cpad = (C + 15) & ~15;

  const int* srcIdx = ei;
  const int* dstIdx = ei + E;

  float* dinv = (float*)d_ws;
  float* bufA = dinv + ws_align((size_t)N);
  float* bufB = bufA + ws_align((size_t)N * (size_t)H);
  float* W3p  = bufB + ws_align((size_t)N * (size_t)H);
  float* b3p  = W3p + ws_align((size_t)H * (size_t)Cpad);

  const int T = 256;

  // ---- Degree normalization: dinv = rsqrt(deg(dst) + 1) ----
  zero_f32<<<1024, T, 0, stream>>>(dinv, (size_t)N);
  deg_count<<<(E + T - 1) / T, T, 0, stream>>>(dstIdx, dinv, E);
  deg_to_dinv<<<(N + T - 1) / T, T, 0, stream>>>(dinv, N);

  // ---- Pad layer-3 weights once (H x C -> H x Cpad) ----
  pad_weights<<<(H * Cpad + T - 1) / T, T, 0, stream>>>(W3, b3, W3p, b3p,
                                                        H, C, Cpad);

  const int tilesM = (N + 15) / 16;
  const int edgeBlocks = (E + 7) / 8;  // 8 edges (8 waves) per 256-thread block

  // ---- Layer 1: bufB = relu(Â (x@W1+b1)) ----
  gemm_wmma_f32<<<dim3(H / 16, tilesM), 32, 0, stream>>>(
      x, W1, b1, bufA, N, F, H, H);
  zero_f32<<<2048, T, 0, stream>>>(bufB, (size_t)N * (size_t)H);
  scatter_edges<<<edgeBlocks, 256, 0, stream>>>(bufA, srcIdx, dstIdx, dinv,
                                                bufB, E, H);
  finalize_selfloop<<<(int)(((size_t)N * H + T - 1) / T), T, 0, stream>>>(
      bufA, dinv, bufB, N, H, 1);

  // ---- Layer 2: bufB = relu(Â (bufB@W2+b2)) ----
  gemm_wmma_f32<<<dim3(H / 16, tilesM), 32, 0, stream>>>(
      bufB, W2, b2, bufA, N, H, H, H);
  zero_f32<<<2048, T, 0, stream>>>(bufB, (size_t)N * (size_t)H);
  scatter_edges<<<edgeBlocks, 256, 0, stream>>>(bufA, srcIdx, dstIdx, dinv,
                                                bufB, E, H);
  finalize_selfloop<<<(int)(((size_t)N * H + T - 1) / T), T, 0, stream>>>(
      bufA, dinv, bufB, N, H, 1);

  // ---- Layer 3: d_out = Â (bufB@W3+b3), no ReLU ----
  float* out = (float*)d_out;
  gemm_wmma_f32<<<dim3(Cpad / 16, tilesM), 32, 0, stream>>>(
      bufB, W3p, b3p, bufA, N, H, C, Cpad);
  zero_f32<<<1024, T, 0, stream>>>(out, (size_t)N * (size_t)C);
  scatter_edges<<<edgeBlocks, 256, 0, stream>>>(bufA, srcIdx, dstIdx, dinv,
                                                out, E, C);
  finalize_selfloop<<<(int)(((size_t)N * C + T - 1) / T), T, 0, stream>>>(
      bufA, dinv, out, N, C, 0);
}